// SynCo_21440476741941
// MI455X (gfx1250) — compile-verified
//
#include <hip/hip_runtime.h>
#include <math.h>

#define TAU 0.5f
#define EPS 1e-8f
#define D 512

// simnet_dist tiling
#define CHUNKS 4            // 16-col chunks processed per A load (64 cols)
#define GROUPS 8            // 8 groups of 64 cols = 512
#define KSLAB 64            // K-slab staged per double-buffer step
#define NSLABS 8            // 512 / KSLAB
#define WP 68               // KSLAB + 4 pad floats: pitch % 64 == 4 -> conflict-free

typedef float v2f __attribute__((ext_vector_type(2)));
typedef float v8f __attribute__((ext_vector_type(8)));

// ---------------- workspace layout (floats) ----------------
// [0,512)      colsum -> init_c
// [512,1024)   ctilde -> c
// [1024]       Zsum
// [1025]       d2min (as int bits)
// [1026]       c_norm (clamped)
// [2048,2048+N) d2 per row

__global__ void init_ws_k(float* ws) {
    int i = blockIdx.x * blockDim.x + threadIdx.x;
    if (i < 1025) ws[i] = 0.0f;
    if (i == 1025) ((int*)ws)[1025] = 0x7F800000; // +inf
    if (i == 1026) ws[i] = 0.0f;
}

__global__ __launch_bounds__(256) void colsum_k(const float* __restrict__ z,
                                                float* __restrict__ colsum, int N) {
    int col = blockIdx.x * 256 + threadIdx.x;          // 0..511
    int r0 = blockIdx.y * 512;
    int r1 = min(r0 + 512, N);
    float s = 0.0f;
    for (int r = r0; r < r1; ++r) s += z[(size_t)r * D + col];
    atomicAdd(&colsum[col], s);
}

__global__ void finish_mean_k(float* ws, int N) {
    int j = blockIdx.x * 256 + threadIdx.x;
    if (j < D) ws[j] *= (1.0f / (float)N);
}

// one wave per row: d2 = ||z_row - c0||^2 ; global min via int atomicMin (d2>=0)
__global__ __launch_bounds__(256) void d2_k(const float* __restrict__ z,
                                            const float* __restrict__ c0,
                                            float* __restrict__ d2,
                                            int* __restrict__ d2min_bits, int N) {
    int lane = threadIdx.x & 31, wave = threadIdx.x >> 5;
    int row = blockIdx.x * 8 + wave;
    if (row >= N) return;
    const float* zr = z + (size_t)row * D;
    float s = 0.0f;
    #pragma unroll
    for (int it = 0; it < 4; ++it) {
        int k = it * 128 + lane * 4;
        float4 v = *(const float4*)(zr + k);
        float4 c = *(const float4*)(c0 + k);
        float a = v.x - c.x, b2 = v.y - c.y, cc = v.z - c.z, dd = v.w - c.w;
        s += a * a + b2 * b2 + cc * cc + dd * dd;
    }
    for (int off = 1; off < 32; off <<= 1) s += __shfl_xor(s, off, 32);
    if (lane == 0) {
        d2[row] = s;
        atomicMin(d2min_bits, __float_as_int(s));
    }
}

// stable softmax weighted accumulation: ctilde += w*z, Zsum += w
__global__ __launch_bounds__(256) void wsum_k(const float* __restrict__ z,
                                              const float* __restrict__ d2,
                                              const int* __restrict__ d2min_bits,
                                              float* __restrict__ ctilde,
                                              float* __restrict__ Zsum, int N) {
    __shared__ float part[D];
    __shared__ float zpart;
    int tid = threadIdx.x, lane = tid & 31, wave = tid >> 5;
    for (int i = tid; i < D; i += 256) part[i] = 0.0f;
    if (tid == 0) zpart = 0.0f;
    __syncthreads();
    float d2min = __int_as_float(*d2min_bits);
    float acc[16];
    #pragma unroll
    for (int q = 0; q < 16; ++q) acc[q] = 0.0f;
    float wsum = 0.0f;
    for (int rr = 0; rr < 8; ++rr) {
        int row = blockIdx.x * 64 + wave * 8 + rr;
        if (row >= N) break;
        float w = expf((d2min - d2[row]) * (1.0f / TAU));
        const float* zr = z + (size_t)row * D;
        #pragma unroll
        for (int it = 0; it < 4; ++it) {
            float4 v = *(const float4*)(zr + it * 128 + lane * 4);
            acc[it * 4 + 0] += w * v.x;
            acc[it * 4 + 1] += w * v.y;
            acc[it * 4 + 2] += w * v.z;
            acc[it * 4 + 3] += w * v.w;
        }
        if (lane == 0) wsum += w;
    }
    #pragma unroll
    for (int it = 0; it < 4; ++it)
        #pragma unroll
        for (int q = 0; q < 4; ++q)
            atomicAdd(&part[it * 128 + lane * 4 + q], acc[it * 4 + q]);
    if (lane == 0) atomicAdd(&zpart, wsum);
    __syncthreads();
    for (int i = tid; i < D; i += 256) atomicAdd(&ctilde[i], part[i]);
    if (tid == 0) atomicAdd(Zsum, zpart);
}

__global__ void finish_c_k(float* ws) {
    __shared__ float red[D];
    int j = threadIdx.x;                       // 512 threads
    float Z = ws[1024];
    float c = ws[512 + j] / Z;
    ws[512 + j] = c;
    red[j] = c * c;
    __syncthreads();
    for (int s = 256; s > 0; s >>= 1) {
        if (j < s) red[j] += red[j + s];
        __syncthreads();
    }
    if (j == 0) {
        float n = sqrtf(red[0]);
        ws[1026] = n < EPS ? EPS : n;
    }
}

// Stage one K-slab of a 64-column W group into LDS via async copy (ASYNCcnt).
// LDS destination offset = low 32 bits of the generic shared-pointer address.
__device__ __forceinline__ void stage_slab_async(const float* __restrict__ W,
                                                 int jb, int kb, float* dst, int tid) {
    const float* wsrc = W + (size_t)(jb * 64) * D + kb;
    #pragma unroll
    for (int s = 0; s < 4; ++s) {
        int i = tid + s * 256;                 // 64 cols * 16 float4 = 1024 items
        int col = i >> 4;
        int k4 = (i & 15) << 2;
        unsigned lds = (unsigned)(uintptr_t)(&dst[col * WP + k4]);
        unsigned long long ga =
            (unsigned long long)(uintptr_t)(wsrc + (size_t)col * D + k4);
        asm volatile("global_load_async_to_lds_b128 %0, %1, off"
                     :: "v"(lds), "v"(ga) : "memory");
    }
}

// Fused GEMM + cosine distance. Wave owns 16 rows; block loops over 8 groups of
// 64 columns; each group accumulates 4 independent 16x16 WMMA tiles over K=512,
// with the W slab double-buffered in LDS via global_load_async_to_lds_b128.
// One A (z) load feeds 4 WMMAs -> 4x A reuse + 4 independent WMMA chains.
__global__ __launch_bounds__(256) void simnet_dist_k(const float* __restrict__ z,
                                                     const float* __restrict__ W,
                                                     const float* __restrict__ b,
                                                     const float* __restrict__ cvec,
                                                     const float* __restrict__ cn_ptr,
                                                     float* __restrict__ out, int N) {
    __shared__ float Wl[2][64 * WP];           // ~35 KB double buffer
    const int tid = threadIdx.x;
    const int lane = tid & 31;
    const int wave = tid >> 5;
    const int m = lane & 15;                   // A row / B col within 16-tile
    const int h = lane >> 4;                   // half-wave: K pair select / M+8
    const int rowbase = (blockIdx.x * 8 + wave) * 16;
    const bool active = rowbase < N;
    const float cn = cn_ptr[0];

    // A-operand base: z[rowbase+m][2h], pairs {K, K+1} per ISA 32-bit A layout
    const float* zp = z + (size_t)(rowbase + m) * D + 2 * h;
    if (active) __builtin_prefetch(zp, 0, 3);  // global_prefetch_b8

    float nrm[8], dot[8];
    #pragma unroll
    for (int r = 0; r < 8; ++r) { nrm[r] = 0.0f; dot[r] = 0.0f; }

    for (int jb = 0; jb < GROUPS; ++jb) {
        v8f acc[CHUNKS];
        #pragma unroll
        for (int q = 0; q < CHUNKS; ++q)
            acc[q] = (v8f){0.f, 0.f, 0.f, 0.f, 0.f, 0.f, 0.f, 0.f};

        // prime the pipeline: slab 0 -> buf 0
        stage_slab_async(W, jb, 0, Wl[0], tid);
        asm volatile("s_wait_asynccnt 0x0" ::: "memory");
        __syncthreads();

        for (int ks = 0; ks < NSLABS; ++ks) {
            if (ks + 1 < NSLABS)               // async prefetch next slab
                stage_slab_async(W, jb, (ks + 1) * KSLAB, Wl[(ks + 1) & 1], tid);

            if (active) {
                const float* wl = &Wl[ks & 1][m * WP + 2 * h];
                const float* zk = zp + ks * KSLAB;
                #pragma unroll 4
                for (int k0 = 0; k0 < KSLAB; k0 += 4) {
                    v2f a = *(const v2f*)(zk + k0);           // one A load ...
                    #pragma unroll
                    for (int q = 0; q < CHUNKS; ++q) {        // ... feeds 4 WMMAs
                        v2f bb = *(const v2f*)(wl + q * 16 * WP + k0);
                        acc[q] = __builtin_amdgcn_wmma_f32_16x16x4_f32(
                                     false, a, false, bb, (short)0, acc[q],
                                     false, false);
                    }
                }
            }
            asm volatile("s_wait_asynccnt 0x0" ::: "memory");
            __syncthreads();
        }

        if (active) {
            #pragma unroll
            for (int q = 0; q < CHUNKS; ++q) {
                int col = jb * 64 + q * 16 + m;
                float bval = b[col];
                float cval = cvec[col];
                #pragma unroll
                for (int r = 0; r < 8; ++r) {  // C layout: row r+8h, col
                    float x = acc[q][r] + bval;
                    nrm[r] += x * x;
                    dot[r] += x * cval;
                }
            }
        }
    }

    if (active) {
        #pragma unroll
        for (int r = 0; r < 8; ++r) {
            float a = nrm[r], d = dot[r];
            for (int off = 1; off < 16; off <<= 1) {  // reduce 16 lanes per half
                a += __shfl_xor(a, off, 32);
                d += __shfl_xor(d, off, 32);
            }
            if (m == r) {
                float xn = sqrtf(a);
                xn = xn < EPS ? EPS : xn;
                out[rowbase + r + 8 * h] = 1.0f - d / (xn * cn);
            }
        }
    }
}

extern "C" void kernel_launch(void* const* d_in, const int* in_sizes, int n_in,
                              void* d_out, int out_size, void* d_ws, size_t ws_size,
                              hipStream_t stream) {
    const float* z = (const float*)d_in[0];
    const float* W = (const float*)d_in[1];
    const float* b = (const float*)d_in[2];
    float* out = (float*)d_out;
    float* ws = (float*)d_ws;
    const int N = in_sizes[0] / D;

    init_ws_k<<<5, 256, 0, stream>>>(ws);
    colsum_k<<<dim3(2, (N + 511) / 512), 256, 0, stream>>>(z, ws, N);
    finish_mean_k<<<2, 256, 0, stream>>>(ws, N);
    d2_k<<<(N + 7) / 8, 256, 0, stream>>>(z, ws, ws + 2048, (int*)ws + 1025, N);
    wsum_k<<<(N + 63) / 64, 256, 0, stream>>>(z, ws + 2048, (const int*)ws + 1025,
                                              ws + 512, ws + 1024, N);
    finish_c_k<<<1, 512, 0, stream>>>(ws);

    int tiles = (N + 15) / 16;
    int blocks = (tiles + 7) / 8;
    simnet_dist_k<<<blocks, 256, 0, stream>>>(z, W, b, ws + 512, ws + 1026, out, N);
}